// LNN_66322884985089
// MI455X (gfx1250) — compile-verified
//
#include <hip/hip_runtime.h>
#include <math.h>

typedef __attribute__((ext_vector_type(2))) float v2f;
typedef __attribute__((ext_vector_type(8))) float v8f;

#define HIDN   128
#define BATCHN 131072
#define GRIDN  1024
#define WAVES  8
#define NSAMP  2                                        // samples per wave in regs
#define ITERS  (BATCHN / (GRIDN * WAVES * NSAMP))       // 8

// LDS layout (floats). Stride 132 => bank-conflict-free b64 fragment loads.
#define WT_STRIDE 132
#define WT_LAYER  (WT_STRIDE * HIDN)                    // 16896
#define STG_BASE  (3 * WT_LAYER)                        // 50688
#define STG_WAVE  (16 * WT_STRIDE)                      // 2112
#define W1_OFF    (STG_BASE + WAVES * STG_WAVE)         // 67584
#define B1_OFF    (W1_OFF + 4 * HIDN)                   // 68096
#define B234_OFF  (B1_OFF + HIDN)                       // 68224
#define W5_OFF    (B234_OFF + 3 * HIDN)                 // 68608
#define SMEM_FLOATS (W5_OFF + HIDN)                     // 68736
#define SMEM_BYTES  (SMEM_FLOATS * 4)                   // 274944 (< 320KB WGP LDS)

// fast device transcendentals (native v_exp_f32 / v_log_f32 / v_rcp_f32)
__device__ __forceinline__ float fast_rcp(float x) { return __builtin_amdgcn_rcpf(x); }

__global__ __launch_bounds__(256) void lnn_el_kernel(
    const float* __restrict__ X,
    const float* __restrict__ W1, const float* __restrict__ B1,
    const float* __restrict__ W2, const float* __restrict__ B2,
    const float* __restrict__ W3, const float* __restrict__ B3,
    const float* __restrict__ W4, const float* __restrict__ B4,
    const float* __restrict__ W5v,
    float* __restrict__ out)
{
  extern __shared__ float smem[];
  const int tid  = threadIdx.x;
  const int wave = tid >> 5;
  const int lane = tid & 31;
  const int nl   = lane & 15;     // column-within-tile / A-row
  const int hi   = lane >> 4;     // half-wave select

  // ---------- cooperative LDS fill: W2..W4 transposed (WT[n][k]) + small params ----------
  {
    const float* Ws[3] = { W2, W3, W4 };
    #pragma unroll 1
    for (int L = 0; L < 3; ++L) {
      const float* W = Ws[L];
      float* WT = smem + L * WT_LAYER;
      for (int i = tid; i < HIDN * HIDN; i += 256) {
        int k = i >> 7, n = i & 127;          // W[k][n] -> WT[n][k]
        WT[n * WT_STRIDE + k] = W[i];
      }
    }
    for (int i = tid; i < 4 * HIDN; i += 256) smem[W1_OFF + i] = W1[i];
    if (tid < HIDN) {
      smem[B1_OFF + tid]            = B1[tid];
      smem[B234_OFF + tid]          = B2[tid];
      smem[B234_OFF + HIDN + tid]   = B3[tid];
      smem[B234_OFF + 2*HIDN + tid] = B4[tid];
      smem[W5_OFF + tid]            = W5v[tid];
    }
  }
  __syncthreads();

  float* stg = smem + STG_BASE + wave * STG_WAVE;   // per-wave private staging

  #pragma unroll 1
  for (int it = 0; it < ITERS; ++it) {
    const int base = (it * GRIDN + (int)blockIdx.x) * (WAVES * NSAMP) + wave * NSAMP;

    // state rows per sample (C-layout, M = r + 8*hi):
    //   low  r0..7 : [v, g0, g1, g2, g3, H20, H21, H22]
    //   high r0..3 : [H23, H30, H31, H33], r4..7 padding (don't care)
    v8f st[NSAMP][8];
    float xv2[NSAMP], xv3[NSAMP];

    // ---------------- layer 1 (4 -> 128), closed form ----------------
    #pragma unroll
    for (int j = 0; j < NSAMP; ++j) {
      const float* xp = X + (size_t)(base + j) * 4;
      float x0 = xp[0], x1 = xp[1], x2 = xp[2], x3 = xp[3];
      xv2[j] = x2; xv3[j] = x3;
      #pragma unroll
      for (int t = 0; t < 8; ++t) {
        int n = t * 16 + nl;
        float w0 = smem[W1_OFF + n];
        float w1 = smem[W1_OFF + HIDN + n];
        float w2 = smem[W1_OFF + 2*HIDN + n];
        float w3 = smem[W1_OFF + 3*HIDN + n];
        float u  = smem[B1_OFF + n];
        u = fmaf(x0, w0, u); u = fmaf(x1, w1, u);
        u = fmaf(x2, w2, u); u = fmaf(x3, w3, u);
        float e   = __expf(-fabsf(u));
        float rr  = fast_rcp(1.0f + e);
        float sig = (u >= 0.0f) ? rr : 1.0f - rr;          // sigmoid(u)
        float sp  = fmaxf(u, 0.0f) + __logf(1.0f + e);     // softplus(u)
        float dd  = sig * (1.0f - sig);                     // softplus''(u)
        v8f c;
        c[0] = hi ? dd*w2*w3 : sp;
        c[1] = hi ? dd*w3*w0 : sig*w0;
        c[2] = hi ? dd*w3*w1 : sig*w1;
        c[3] = hi ? dd*w3*w3 : sig*w2;
        c[4] = hi ? 0.0f     : sig*w3;
        c[5] = hi ? 0.0f     : dd*w2*w0;
        c[6] = hi ? 0.0f     : dd*w2*w1;
        c[7] = hi ? 0.0f     : dd*w2*w2;
        st[j][t] = c;
      }
    }

    // ---------------- layers 2..4: WMMA GEMM + chain-rule nonlinearity ----------------
    #pragma unroll 1
    for (int L = 0; L < 3; ++L) {
      const float* WT   = smem + L * WT_LAYER;
      const float* bias = smem + B234_OFF + L * HIDN;
      #pragma unroll
      for (int j = 0; j < NSAMP; ++j) {
        // stage state rows 0..11 (M-major, k contiguous) for A-fragment reads
        #pragma unroll
        for (int t = 0; t < 8; ++t)
          #pragma unroll
          for (int r = 0; r < 8; ++r)
            stg[(r + 8*hi) * WT_STRIDE + t*16 + nl] = st[j][t][r];

        // accumulators: 8 N-tiles resident; bias only into v row (M=0, low half VGPR0)
        v8f acc[8];
        #pragma unroll
        for (int t = 0; t < 8; ++t) {
          v8f z = {};
          z[0] = hi ? 0.0f : bias[t*16 + nl];
          acc[t] = z;
        }

        // k-outer / tile-inner: one A fragment live, 8 independent WMMA chains
        const float* abase = &stg[nl * WT_STRIDE + 2*hi];
        const float* bbase = WT + nl * WT_STRIDE + 2*hi;
        #pragma unroll
        for (int c = 0; c < 32; ++c) {
          v2f a = *(const v2f*)&abase[c*4];       // A: row m=nl, K=4c+2hi+{0,1}
          #pragma unroll
          for (int t = 0; t < 8; ++t) {
            v2f b = *(const v2f*)&bbase[c*4 + t*16*WT_STRIDE];  // B: N=16t+nl
            acc[t] = __builtin_amdgcn_wmma_f32_16x16x4_f32(
                false, a, false, b, (short)0, acc[t], false, false);
          }
        }

        // nonlinearity per tile: high half needs u,g0..g3 from partner low lane
        #pragma unroll
        for (int t = 0; t < 8; ++t) {
          v8f ac = acc[t];
          float o0 = __shfl_xor(ac[0], 16);
          float o1 = __shfl_xor(ac[1], 16);
          float o2 = __shfl_xor(ac[2], 16);
          float o3 = __shfl_xor(ac[3], 16);
          float o4 = __shfl_xor(ac[4], 16);
          float u  = hi ? o0 : ac[0];
          float g0 = hi ? o1 : ac[1];
          float g1 = hi ? o2 : ac[2];
          float g2 = hi ? o3 : ac[3];
          float g3 = hi ? o4 : ac[4];
          float e   = __expf(-fabsf(u));
          float rr  = fast_rcp(1.0f + e);
          float sig = (u >= 0.0f) ? rr : 1.0f - rr;
          float sp  = fmaxf(u, 0.0f) + __logf(1.0f + e);
          float dd  = sig * (1.0f - sig);
          v8f nc;
          nc[0] = hi ? fmaf(dd*g2, g3, sig*ac[0]) : sp;
          nc[1] = hi ? fmaf(dd*g3, g0, sig*ac[1]) : sig*g0;
          nc[2] = hi ? fmaf(dd*g3, g1, sig*ac[2]) : sig*g1;
          nc[3] = hi ? fmaf(dd*g3, g3, sig*ac[3]) : sig*g2;
          nc[4] = hi ? 0.0f : sig*g3;
          nc[5] = hi ? 0.0f : fmaf(dd*g2, g0, sig*ac[5]);
          nc[6] = hi ? 0.0f : fmaf(dd*g2, g1, sig*ac[6]);
          nc[7] = hi ? 0.0f : fmaf(dd*g2, g2, sig*ac[7]);
          st[j][t] = nc;
        }
      }
    }

    // ---------------- layer 5 dot(W5) + 2x2 pinv solve ----------------
    #pragma unroll
    for (int j = 0; j < NSAMP; ++j) {
      float p[8] = {0,0,0,0,0,0,0,0};
      #pragma unroll
      for (int t = 0; t < 8; ++t) {
        float w5 = smem[W5_OFF + t*16 + nl];
        #pragma unroll
        for (int r = 0; r < 8; ++r) p[r] = fmaf(st[j][t][r], w5, p[r]);
      }
      #pragma unroll
      for (int m = 1; m <= 8; m <<= 1)
        #pragma unroll
        for (int r = 0; r < 8; ++r) p[r] += __shfl_xor(p[r], m);
      float q[8];
      #pragma unroll
      for (int r = 0; r < 8; ++r) q[r] = __shfl_xor(p[r], 16);
      // D[0..7] from low half, D[8..11] from high half
      float J0  = hi ? q[1] : p[1];
      float J1  = hi ? q[2] : p[2];
      float H20 = hi ? q[5] : p[5];
      float H21 = hi ? q[6] : p[6];
      float H22 = hi ? q[7] : p[7];
      float H23 = hi ? p[0] : q[0];
      float H30 = hi ? p[1] : q[1];
      float H31 = hi ? p[2] : q[2];
      float H33 = hi ? p[3] : q[3];

      float r0 = J0 - H20*xv2[j] - H21*xv3[j];
      float r1 = J1 - H30*xv2[j] - H31*xv3[j];

      // pinv of symmetric [[a,b],[b,c]] via eigendecomposition + sv cutoff
      float aa = H22, bb = H23, cc = H33;
      float tr = aa + cc, df = aa - cc;
      float sd = sqrtf(fmaf(df, df, 4.0f*bb*bb));
      float l1 = 0.5f*(tr + sd), l2 = 0.5f*(tr - sd);
      float mx  = fmaxf(fabsf(l1), fabsf(l2));
      float cut = 4.7683716e-7f * mx;                      // ~2*2*eps, pinv-style rcond
      float il1 = (fabsf(l1) > cut) ? 1.0f/l1 : 0.0f;
      float il2 = (fabsf(l2) > cut) ? 1.0f/l2 : 0.0f;
      float vx = bb, vy = l1 - aa;
      float nn = fmaf(vx, vx, vy*vy);
      int   dg = (nn < 1e-30f);
      vx = dg ? 1.0f : vx; vy = dg ? 0.0f : vy; nn = dg ? 1.0f : nn;
      float inv = 1.0f / nn;
      float p00 = (il1*vx*vx + il2*vy*vy) * inv;
      float p01 = (il1 - il2) * vx * vy * inv;
      float p11 = (il1*vy*vy + il2*vx*vx) * inv;
      float out0 = p00*r0 + p01*r1;
      float out1 = p01*r0 + p11*r1;

      if (lane == 0) {
        size_t smp = (size_t)(base + j);
        out[smp*2 + 0] = out0;
        out[smp*2 + 1] = out1;
      }
    }
  }
}

extern "C" void kernel_launch(void* const* d_in, const int* in_sizes, int n_in,
                              void* d_out, int out_size, void* d_ws, size_t ws_size,
                              hipStream_t stream) {
  (void)in_sizes; (void)n_in; (void)out_size; (void)d_ws; (void)ws_size;
  const float* X  = (const float*)d_in[0];
  const float* W1 = (const float*)d_in[1];
  const float* B1 = (const float*)d_in[2];
  const float* W2 = (const float*)d_in[3];
  const float* B2 = (const float*)d_in[4];
  const float* W3 = (const float*)d_in[5];
  const float* B3 = (const float*)d_in[6];
  const float* W4 = (const float*)d_in[7];
  const float* B4 = (const float*)d_in[8];
  const float* W5 = (const float*)d_in[9];
  float* out = (float*)d_out;

  (void)hipFuncSetAttribute((const void*)lnn_el_kernel,
                            hipFuncAttributeMaxDynamicSharedMemorySize, SMEM_BYTES);
  lnn_el_kernel<<<GRIDN, 256, SMEM_BYTES, stream>>>(X, W1, B1, W2, B2, W3, B3,
                                                    W4, B4, W5, out);
}